// RGCN_7739531067737
// MI455X (gfx1250) — compile-verified
//
#include <hip/hip_runtime.h>

#define D 256
#define NGENE 4762
#define NCELL 847
#define KSTEPS (D / 4)   // 64 WMMA k-steps
#define MT_PER 4         // m-tiles processed per wave (B-frag register reuse)

typedef __attribute__((ext_vector_type(2))) float v2f;
typedef __attribute__((ext_vector_type(8))) float v8f;

// ---------------- utility: zero workspace ----------------
__global__ void zero_ws_kernel(float* __restrict__ ws, int n) {
  for (int i = blockIdx.x * blockDim.x + threadIdx.x; i < n;
       i += gridDim.x * blockDim.x)
    ws[i] = 0.0f;
}

// ---------------- degree counts (segment_sum of ones) ----------------
__global__ void degree_kernel(const int* __restrict__ src, const int* __restrict__ dst,
                              float* __restrict__ deg_src, float* __restrict__ deg_dst,
                              int E) {
  int e = blockIdx.x * blockDim.x + threadIdx.x;
  if (e < E) {
    atomicAdd(&deg_src[src[e]], 1.0f);
    atomicAdd(&deg_dst[dst[e]], 1.0f);
  }
}

// deg -> clip(deg,1)^-0.5 in place
__global__ void finalize_norm_kernel(float* __restrict__ deg, int n) {
  int i = blockIdx.x * blockDim.x + threadIdx.x;
  if (i < n) {
    float d = deg[i];
    deg[i] = rsqrtf(d < 1.0f ? 1.0f : d);
  }
}

// ---------------- edge scatter-add: agg[dst] += x[src] * norm_src[src] ----------------
// grid.x = E, block = D threads. agg arrays (<5 MB) are L2-resident, so these
// atomics stay on-chip.
__global__ void scatter_kernel(const float* __restrict__ x,
                               const float* __restrict__ norm_src,
                               const int* __restrict__ src, const int* __restrict__ dst,
                               float* __restrict__ agg, int E) {
  int e = blockIdx.x;
  if (e >= E) return;
  int s = src[e];
  int d = dst[e];
  int f = threadIdx.x;
  float v = x[s * D + f] * norm_src[s];
  atomicAdd(&agg[d * D + f], v);
}

// ---------------- WMMA fp32 GEMM: Y = relu((X * norm_dst[:,None]) @ W + b) --------------
// One wave owns one 16-wide n-tile; it preloads ALL 64 B-fragments of that
// n-tile into registers (64 x v2f = 128 VGPRs), then sweeps MT_PER m-tiles
// reusing them. Tail m-rows are handled branchlessly: load row clamped to M-1
// and nrm forced to 0, so the K-loop has zero divergence and the compiler can
// clause the 8-byte A loads and pipeline the WMMAs.
// Layouts (wave32):
//   A (16x4 f32):  lane -> M = lane&15, K = k0 + (lane>>4)*2 + {0,1}
//   B (4x16 f32):  lane -> N = lane&15, same K split
//   C/D (16x16):   8 VGPRs, M = vgpr + (lane>=16 ? 8:0), N = lane&15
__global__ void __launch_bounds__(128)
gemm_wmma_kernel(const float* __restrict__ X,
                 const float* __restrict__ norm_dst,
                 const float* __restrict__ W,
                 const float* __restrict__ bias,
                 float* __restrict__ Y, int M, int mtiles) {
  const int lane = threadIdx.x;                    // 0..31
  const int nt   = blockIdx.y * 4 + threadIdx.y;   // n-tile 0..15
  const int ncol = nt * 16 + (lane & 15);
  const int kg   = (lane >> 4) * 2;                // 0 or 2

  // Preload B fragments for this n-tile (reused across all m-tiles).
  v2f bfrag[KSTEPS];
#pragma unroll
  for (int i = 0; i < KSTEPS; ++i) {
    const int ka = i * 4 + kg;
    bfrag[i].x = W[ka * D + ncol];
    bfrag[i].y = W[(ka + 1) * D + ncol];
  }
  const float bb = bias[ncol];

  const int mt0    = blockIdx.x * MT_PER;
  const int mt_end = (mt0 + MT_PER < mtiles) ? (mt0 + MT_PER) : mtiles;

  for (int mt = mt0; mt < mt_end; ++mt) {
    const int mrow  = mt * 16 + (lane & 15);
    const int mload = (mrow < M) ? mrow : (M - 1);          // clamp: always legal
    const float nrm = (mrow < M) ? norm_dst[mload] : 0.0f;  // zero pad rows
    const float2* xrow = (const float2*)(X + (size_t)mload * D);

    v8f acc = {};
#pragma unroll
    for (int i = 0; i < KSTEPS; ++i) {
      const float2 ax = xrow[(i * 4 + kg) >> 1];   // one aligned 8B load
      v2f a;
      a.x = ax.x * nrm;
      a.y = ax.y * nrm;
      acc = __builtin_amdgcn_wmma_f32_16x16x4_f32(false, a, false, bfrag[i],
                                                  (short)0, acc, false, false);
    }

    const int mbase = mt * 16 + ((lane >> 4) ? 8 : 0);
#pragma unroll
    for (int i = 0; i < 8; ++i) {
      const int r = mbase + i;
      if (r < M) {
        float v = acc[i] + bb;
        Y[(size_t)r * D + ncol] = v > 0.0f ? v : 0.0f;
      }
    }
  }
}

// ---------------- predictor: score[e] = concat(hg[src], hc[dst]) . Wp + bp ----------------
__global__ void predict_kernel(const float* __restrict__ hg, const float* __restrict__ hc,
                               const float* __restrict__ Wp, const float* __restrict__ bp,
                               const int* __restrict__ src, const int* __restrict__ dst,
                               float* __restrict__ score, int E) {
  int e = blockIdx.x * blockDim.x + threadIdx.x;
  if (e >= E) return;
  const float4* g  = (const float4*)(hg + (size_t)src[e] * D);
  const float4* c  = (const float4*)(hc + (size_t)dst[e] * D);
  const float4* w0 = (const float4*)Wp;
  const float4* w1 = (const float4*)(Wp + D);
  float acc = bp[0];
#pragma unroll 8
  for (int i = 0; i < D / 4; ++i) {
    float4 a = g[i], w = w0[i];
    acc += a.x * w.x + a.y * w.y + a.z * w.z + a.w * w.w;
  }
#pragma unroll 8
  for (int i = 0; i < D / 4; ++i) {
    float4 a = c[i], w = w1[i];
    acc += a.x * w.x + a.y * w.y + a.z * w.z + a.w * w.w;
  }
  score[e] = acc;
}

extern "C" void kernel_launch(void* const* d_in, const int* in_sizes, int n_in,
                              void* d_out, int out_size, void* d_ws, size_t ws_size,
                              hipStream_t stream) {
  // inputs in setup_inputs() order
  const float* gene_emb = (const float*)d_in[0];
  const float* cell_emb = (const float*)d_in[1];
  const float* W_g2c    = (const float*)d_in[2];
  const float* b_g2c    = (const float*)d_in[3];
  const float* W_c2g    = (const float*)d_in[4];
  const float* b_c2g    = (const float*)d_in[5];
  const float* Wp       = (const float*)d_in[6];
  const float* bp       = (const float*)d_in[7];
  const int* g2c_src    = (const int*)d_in[8];
  const int* g2c_dst    = (const int*)d_in[9];
  const int* c2g_src    = (const int*)d_in[10];
  const int* c2g_dst    = (const int*)d_in[11];
  const int* dec_src    = (const int*)d_in[12];
  const int* dec_dst    = (const int*)d_in[13];

  const int E    = in_sizes[8];   // 200000
  const int Edec = in_sizes[12];  // 200000

  // workspace layout (floats)
  float* ws       = (float*)d_ws;
  float* deg_go   = ws;                    // gene out-degree  (g2c src)  [NGENE]
  float* deg_ci   = deg_go + NGENE;        // cell in-degree   (g2c dst)  [NCELL]
  float* deg_co   = deg_ci + NCELL;        // cell out-degree  (c2g src)  [NCELL]
  float* deg_gi   = deg_co + NCELL;        // gene in-degree   (c2g dst)  [NGENE]
  float* agg_cell = deg_gi + NGENE;        // [NCELL, D]
  float* agg_gene = agg_cell + NCELL * D;  // [NGENE, D]
  const int n_norm    = 2 * (NGENE + NCELL);
  const int ws_floats = n_norm + (NCELL + NGENE) * D;

  // outputs concatenated: score [Edec], h_gene [NGENE*D], h_cell [NCELL*D]
  float* score  = (float*)d_out;
  float* h_gene = score + Edec;
  float* h_cell = h_gene + NGENE * D;

  // 1) zero degree + agg workspace
  zero_ws_kernel<<<512, 256, 0, stream>>>(ws, ws_floats);

  // 2) degrees
  degree_kernel<<<(E + 255) / 256, 256, 0, stream>>>(g2c_src, g2c_dst, deg_go, deg_ci, E);
  degree_kernel<<<(E + 255) / 256, 256, 0, stream>>>(c2g_src, c2g_dst, deg_co, deg_gi, E);

  // 3) deg -> clip(deg,1)^-0.5 (all four arrays are contiguous)
  finalize_norm_kernel<<<(n_norm + 255) / 256, 256, 0, stream>>>(ws, n_norm);

  // 4) edge scatter-add with src normalization fused
  scatter_kernel<<<E, D, 0, stream>>>(gene_emb, deg_go, g2c_src, g2c_dst, agg_cell, E);
  scatter_kernel<<<E, D, 0, stream>>>(cell_emb, deg_co, c2g_src, c2g_dst, agg_gene, E);

  // 5) WMMA GEMM + bias + ReLU, dst normalization fused (writes straight to d_out)
  {
    dim3 blk(32, 4);
    const int mt_c = (NCELL + 15) / 16;   // 53
    const int mt_g = (NGENE + 15) / 16;   // 298
    dim3 grd_c((mt_c + MT_PER - 1) / MT_PER, 4);
    gemm_wmma_kernel<<<grd_c, blk, 0, stream>>>(agg_cell, deg_ci, W_g2c, b_g2c,
                                                h_cell, NCELL, mt_c);
    dim3 grd_g((mt_g + MT_PER - 1) / MT_PER, 4);
    gemm_wmma_kernel<<<grd_g, blk, 0, stream>>>(agg_gene, deg_gi, W_c2g, b_c2g,
                                                h_gene, NGENE, mt_g);
  }

  // 6) edge predictor
  predict_kernel<<<(Edec + 255) / 256, 256, 0, stream>>>(h_gene, h_cell, Wp, bp,
                                                         dec_src, dec_dst, score, Edec);
}